// EmbeddingM_17755394802311
// MI455X (gfx1250) — compile-verified
//
#include <hip/hip_runtime.h>
#include <math.h>

#define N_NODES 16384
#define FMDIM   256
#define N_EDGES 524288
#define NF      (N_NODES * FMDIM)   // 4,194,304 floats per feature matrix

typedef float v2f __attribute__((ext_vector_type(2)));
typedef float v8f __attribute__((ext_vector_type(8)));

// ---------------------------------------------------------------------------
// GEMM: H = A(16384x256) @ B(256x256), f32, via V_WMMA_F32_16X16X4_F32.
// One wave per 16x16 output tile; K looped in steps of 4 (64 WMMAs per tile).
// A-frag layout: lanes 0-15 rows M=0..15 with K=k,k+1; lanes 16-31 K=k+2,k+3.
// B-frag layout (mirror): lanes 0-15 cols N=0..15 rows K=k,k+1; lanes 16-31 K=k+2,k+3.
// C/D layout (ISA 7.12.2): VGPR r, lanes 0-15 -> M=r, lanes 16-31 -> M=r+8.
// ---------------------------------------------------------------------------
__global__ void gcn_gemm(const float* __restrict__ A, const float* __restrict__ B,
                         float* __restrict__ C) {
    const int wave = threadIdx.x >> 5;               // 4 waves per block
    const int lane = threadIdx.x & 31;
    const int tile = blockIdx.x * 4 + wave;          // 0 .. 16383
    const int tm   = tile >> 4;                      // 1024 M-tiles
    const int tn   = tile & 15;                      // 16 N-tiles
    const int m0 = tm << 4, n0 = tn << 4;
    const int l15   = lane & 15;
    const int khalf = (lane >> 4) << 1;              // 0 or 2

    const float* __restrict__ arow = A + (size_t)(m0 + l15) * FMDIM + khalf;
    const float* __restrict__ bcol = B + (size_t)khalf * FMDIM + (n0 + l15);

    v8f c = {};
#pragma unroll 8
    for (int k = 0; k < FMDIM; k += 4) {
        v2f a, b;
        a.x = arow[k];
        a.y = arow[k + 1];
        b.x = bcol[(size_t)k * FMDIM];
        b.y = bcol[(size_t)(k + 1) * FMDIM];
        c = __builtin_amdgcn_wmma_f32_16x16x4_f32(
                /*neg_a=*/false, a, /*neg_b=*/false, b,
                /*c_mod=*/(short)0, c, /*reuse_a=*/false, /*reuse_b=*/false);
    }

    const int rbase = m0 + ((lane >> 4) << 3);       // lanes>=16 hold M = r+8
    float* __restrict__ cp = C + (n0 + l15);
#pragma unroll
    for (int r = 0; r < 8; ++r)
        cp[(size_t)(rbase + r) * FMDIM] = c[r];
}

// ---------------------------------------------------------------------------
// Degree / norm helpers
// ---------------------------------------------------------------------------
__global__ void init_deg(float* __restrict__ deg) {
    int i = blockIdx.x * blockDim.x + threadIdx.x;
    if (i < N_NODES) deg[i] = 1.0f;                  // self-loop weight
}

__global__ void deg_accum(const int* __restrict__ dst, const float* __restrict__ ew,
                          float* __restrict__ deg, int E) {
    int stride = gridDim.x * blockDim.x;
    for (int e = blockIdx.x * blockDim.x + threadIdx.x; e < E; e += stride)
        atomicAdd(&deg[dst[e]], ew[e]);
}

__global__ void dinv_kernel(const float* __restrict__ deg, float* __restrict__ dinv) {
    int i = blockIdx.x * blockDim.x + threadIdx.x;
    if (i < N_NODES) {
        float d = deg[i];
        dinv[i] = (d > 0.0f) ? rsqrtf(fmaxf(d, 1e-30f)) : 0.0f;
    }
}

// out[n,f] = b[f] + dinv[n]^2 * H[n,f]   (bias + self-loop message)
__global__ void init_out(const float* __restrict__ H, const float* __restrict__ dinv,
                         const float* __restrict__ bias, float* __restrict__ out) {
    int i = blockIdx.x * blockDim.x + threadIdx.x;
    if (i < NF) {
        int n = i >> 8, f = i & 255;
        float di = dinv[n];
        out[i] = bias[f] + di * di * H[i];
    }
}

// One wave per edge (grid-stride). Each lane moves 8 features (two float4 loads,
// eight global f32 atomic adds -> global_atomic_add_f32, L2-resident).
__global__ void gcn_scatter(const float* __restrict__ H, const int* __restrict__ src,
                            const int* __restrict__ dst, const float* __restrict__ ew,
                            const float* __restrict__ dinv, float* __restrict__ out,
                            int E) {
    const int lane   = threadIdx.x & 31;
    const int warp   = (blockIdx.x * blockDim.x + threadIdx.x) >> 5;
    const int nwarps = (gridDim.x * blockDim.x) >> 5;
    for (int e = warp; e < E; e += nwarps) {
        const int   s    = src[e];
        const int   d    = dst[e];
        const float norm = dinv[s] * ew[e] * dinv[d];
        const float4 h0 = *(const float4*)(H + (size_t)s * FMDIM + lane * 8);
        const float4 h1 = *(const float4*)(H + (size_t)s * FMDIM + lane * 8 + 4);
        float* o = out + (size_t)d * FMDIM + lane * 8;
        atomicAdd(o + 0, norm * h0.x);
        atomicAdd(o + 1, norm * h0.y);
        atomicAdd(o + 2, norm * h0.z);
        atomicAdd(o + 3, norm * h0.w);
        atomicAdd(o + 4, norm * h1.x);
        atomicAdd(o + 5, norm * h1.y);
        atomicAdd(o + 6, norm * h1.z);
        atomicAdd(o + 7, norm * h1.w);
    }
}

__global__ void relu_kernel(float* __restrict__ x) {
    int i = blockIdx.x * blockDim.x + threadIdx.x;
    if (i < NF) x[i] = fmaxf(x[i], 0.0f);
}

// ---------------------------------------------------------------------------
// SE channel attention
// ---------------------------------------------------------------------------
__global__ void zero_small(float* __restrict__ p, int n) {
    if (threadIdx.x < n) p[threadIdx.x] = 0.0f;
}

__global__ void pool_sum(const float* __restrict__ branch, float* __restrict__ pooled) {
    const int c = blockIdx.y;
    const float* __restrict__ p = branch + (size_t)c * NF;
    float acc = 0.0f;
    for (int i = blockIdx.x * blockDim.x + threadIdx.x; i < NF;
         i += gridDim.x * blockDim.x)
        acc += p[i];
    for (int o = 16; o > 0; o >>= 1) acc += __shfl_down(acc, o, 32);
    __shared__ float s[8];
    if ((threadIdx.x & 31) == 0) s[threadIdx.x >> 5] = acc;
    __syncthreads();
    if (threadIdx.x == 0) {
        float v = 0.0f;
        for (int w = 0; w < (int)(blockDim.x >> 5); ++w) v += s[w];
        atomicAdd(&pooled[c], v);
    }
}

__global__ void se_kernel(const float* __restrict__ pooled,
                          const float* __restrict__ fc1_w, const float* __restrict__ fc1_b,
                          const float* __restrict__ fc2_w, const float* __restrict__ fc2_b,
                          float* __restrict__ att) {
    __shared__ float p[6];
    __shared__ float h[30];
    const int t = threadIdx.x;
    if (t < 6) p[t] = pooled[t] * (1.0f / ((float)NF));
    __syncthreads();
    if (t < 30) {
        float acc = fc1_b[t];
        for (int c = 0; c < 6; ++c) acc += fc1_w[t * 6 + c] * p[c];
        h[t] = fmaxf(acc, 0.0f);
    }
    __syncthreads();
    if (t < 6) {
        float acc = fc2_b[t];
        for (int j = 0; j < 30; ++j) acc += fc2_w[t * 30 + j] * h[j];
        att[t] = 1.0f / (1.0f + expf(-acc));
    }
}

// out[n,f] = sum_c cnn_w[c] * relu(att[c] * branch_c[n,f]) + cnn_b
__global__ void final_kernel(const float* __restrict__ branch,
                             const float* __restrict__ att,
                             const float* __restrict__ cnn_w,
                             const float* __restrict__ cnn_b,
                             float* __restrict__ out) {
    int i = blockIdx.x * blockDim.x + threadIdx.x;
    if (i >= NF) return;
    float a0 = att[0], a1 = att[1], a2 = att[2], a3 = att[3], a4 = att[4], a5 = att[5];
    float acc = cnn_b[0];
    acc += cnn_w[0] * fmaxf(a0 * branch[(size_t)0 * NF + i], 0.0f);
    acc += cnn_w[1] * fmaxf(a1 * branch[(size_t)1 * NF + i], 0.0f);
    acc += cnn_w[2] * fmaxf(a2 * branch[(size_t)2 * NF + i], 0.0f);
    acc += cnn_w[3] * fmaxf(a3 * branch[(size_t)3 * NF + i], 0.0f);
    acc += cnn_w[4] * fmaxf(a4 * branch[(size_t)4 * NF + i], 0.0f);
    acc += cnn_w[5] * fmaxf(a5 * branch[(size_t)5 * NF + i], 0.0f);
    out[i] = acc;
}

// ---------------------------------------------------------------------------
// Host orchestration
// ---------------------------------------------------------------------------
extern "C" void kernel_launch(void* const* d_in, const int* in_sizes, int n_in,
                              void* d_out, int out_size, void* d_ws, size_t ws_size,
                              hipStream_t stream) {
    (void)in_sizes; (void)n_in; (void)out_size; (void)ws_size;

    const float* x = (const float*)d_in[0];
    const int*   edges[3] = {(const int*)d_in[1], (const int*)d_in[3], (const int*)d_in[5]};
    const float* ews[3]   = {(const float*)d_in[2], (const float*)d_in[4], (const float*)d_in[6]};
    const float* Wt[6]; const float* bt[6];
    for (int i = 0; i < 6; ++i) {
        Wt[i] = (const float*)d_in[7 + 2 * i];
        bt[i] = (const float*)d_in[8 + 2 * i];
    }
    const float* fc1_w = (const float*)d_in[19];
    const float* fc1_b = (const float*)d_in[20];
    const float* fc2_w = (const float*)d_in[21];
    const float* fc2_b = (const float*)d_in[22];
    const float* cnn_w = (const float*)d_in[23];
    const float* cnn_b = (const float*)d_in[24];

    // Workspace layout (floats): 6 branch buffers | H | deg | dinv | pooled | att
    // total ~= 7*NF + 2*N_NODES + 16 floats ~= 118 MB
    float* ws     = (float*)d_ws;
    float* branch = ws;
    float* H      = ws + (size_t)6 * NF;
    float* deg    = H + NF;
    float* dinv   = deg + N_NODES;
    float* pooled = dinv + N_NODES;
    float* att    = pooled + 8;

    const int EBLK = 4096;     // scatter grid
    const int NFBLK = NF / 256;

    for (int br = 0; br < 3; ++br) {
        const int* src = edges[br];
        const int* dst = edges[br] + N_EDGES;
        const float* ew = ews[br];

        init_deg<<<N_NODES / 256, 256, 0, stream>>>(deg);
        deg_accum<<<2048, 256, 0, stream>>>(dst, ew, deg, N_EDGES);
        dinv_kernel<<<N_NODES / 256, 256, 0, stream>>>(deg, dinv);

        const float* in = x;
        for (int l = 0; l < 2; ++l) {
            const int ch = 2 * br + l;
            float* outb = branch + (size_t)ch * NF;
            gcn_gemm<<<4096, 128, 0, stream>>>(in, Wt[ch], H);
            init_out<<<NFBLK, 256, 0, stream>>>(H, dinv, bt[ch], outb);
            gcn_scatter<<<EBLK, 256, 0, stream>>>(H, src, dst, ew, dinv, outb, N_EDGES);
            relu_kernel<<<NFBLK, 256, 0, stream>>>(outb);
            in = outb;
        }
    }

    zero_small<<<1, 32, 0, stream>>>(pooled, 8);
    pool_sum<<<dim3(256, 6), 256, 0, stream>>>(branch, pooled);
    se_kernel<<<1, 32, 0, stream>>>(pooled, fc1_w, fc1_b, fc2_w, fc2_b, att);
    final_kernel<<<NFBLK, 256, 0, stream>>>(branch, att, cnn_w, cnn_b, (float*)d_out);
}